// CAMD_73229192397362
// MI455X (gfx1250) — compile-verified
//
#include <hip/hip_runtime.h>
#include <hip/hip_bf16.h>

typedef float v2f __attribute__((ext_vector_type(2)));
typedef float v8f __attribute__((ext_vector_type(8)));

#define T 8192
#define D 32
#define CS 128
#define NCHUNK (T / CS)

// ---------------------------------------------------------------------------
// Kernel 1: fused 3-layer MLP via fp32 WMMA (V_WMMA_F32_16X16X4_F32).
// One wave (32 threads) computes one 16-row tile of one of 5 jobs:
//   job 0: Q = mlp(m1, Wq, bq);  jobs 1..4: K_m = mlp(m_i, Wk, bk)
// ---------------------------------------------------------------------------
__global__ __launch_bounds__(32) void mlp_wmma_kernel(
    const float* __restrict__ m1, const float* __restrict__ m2,
    const float* __restrict__ m3, const float* __restrict__ m4,
    const float* __restrict__ Wq, const float* __restrict__ bq,
    const float* __restrict__ Wk, const float* __restrict__ bk,
    float* __restrict__ Qout, float* __restrict__ Kout)
{
    __shared__ float lds[16 * D];

    const int lane  = threadIdx.x;   // 0..31 (wave32)
    const int mrow  = lane & 15;     // A-matrix row owned by this lane
    const int khalf = lane >> 4;     // 0: K pair {0,1}, 1: K pair {2,3}
    const int n0    = lane & 15;     // B/C column within half

    const int tile = blockIdx.x;
    const int job  = blockIdx.y;
    const int row0 = tile * 16;

    const float* x; const float* W; const float* B; float* out;
    switch (job) {
        case 0:  x = m1; W = Wq; B = bq; out = Qout;             break;
        case 1:  x = m1; W = Wk; B = bk; out = Kout + 0*T*D;     break;
        case 2:  x = m2; W = Wk; B = bk; out = Kout + 1*T*D;     break;
        case 3:  x = m3; W = Wk; B = bk; out = Kout + 2*T*D;     break;
        default: x = m4; W = Wk; B = bk; out = Kout + 3*T*D;     break;
    }

    // Load 16x32 input tile in the f32 16x16x4 A layout:
    // lane l holds rows m=l&15, K elements {4kc+2*khalf, 4kc+2*khalf+1}.
    v2f a[8];
    const float* xrow = x + (size_t)(row0 + mrow) * D;
    #pragma unroll
    for (int kc = 0; kc < 8; ++kc)
        a[kc] = *(const v2f*)(xrow + 4 * kc + 2 * khalf);

    for (int layer = 0; layer < 3; ++layer) {
        const float* Wl = W + layer * D * D;
        const float* bl = B + layer * D;

        // B layout (4x16 per step): lane l holds column n=l&15,
        // K rows {2*khalf, 2*khalf+1} of each 4-row K slab.
        v2f b0[8], b1[8];
        #pragma unroll
        for (int kc = 0; kc < 8; ++kc) {
            const int kb = 4 * kc + 2 * khalf;
            b0[kc].x = Wl[(kb + 0) * D + n0];
            b0[kc].y = Wl[(kb + 1) * D + n0];
            b1[kc].x = Wl[(kb + 0) * D + n0 + 16];
            b1[kc].y = Wl[(kb + 1) * D + n0 + 16];
        }

        // Seed accumulators with bias (C[m][n] = b[n] for all m).
        const float bias0 = bl[n0];
        const float bias1 = bl[n0 + 16];
        v8f c0, c1;
        #pragma unroll
        for (int v = 0; v < 8; ++v) { c0[v] = bias0; c1[v] = bias1; }

        #pragma unroll
        for (int kc = 0; kc < 8; ++kc) {
            c0 = __builtin_amdgcn_wmma_f32_16x16x4_f32(
                false, a[kc], false, b0[kc], (short)0, c0, false, false);
            c1 = __builtin_amdgcn_wmma_f32_16x16x4_f32(
                false, a[kc], false, b1[kc], (short)0, c1, false, false);
        }

        if (layer < 2) {
            #pragma unroll
            for (int v = 0; v < 8; ++v) {
                c0[v] = fmaxf(c0[v], 0.0f);
                c1[v] = fmaxf(c1[v], 0.0f);
            }
            // C layout -> row-major LDS: lane owns rows v+8*khalf, col n0 / n0+16.
            #pragma unroll
            for (int v = 0; v < 8; ++v) {
                const int rm = v + 8 * khalf;
                lds[rm * D + n0]      = c0[v];
                lds[rm * D + n0 + 16] = c1[v];
            }
            __syncthreads();
            #pragma unroll
            for (int kc = 0; kc < 8; ++kc)
                a[kc] = *(const v2f*)(&lds[mrow * D + 4 * kc + 2 * khalf]);
            __syncthreads();
        } else {
            #pragma unroll
            for (int v = 0; v < 8; ++v) {
                const int rm = v + 8 * khalf;
                out[(size_t)(row0 + rm) * D + n0]      = c0[v];
                out[(size_t)(row0 + rm) * D + n0 + 16] = c1[v];
            }
        }
    }
}

// ---------------------------------------------------------------------------
// Kernel 2: within-chunk inclusive prefix scan of outer[s] = K[s][d]*V[s][c].
// Thread j = c*32+d owns one of 64 scan components; 128 sequential steps.
// ---------------------------------------------------------------------------
__global__ __launch_bounds__(64) void scan_chunk_kernel(
    const float* __restrict__ K,
    const float* __restrict__ m1, const float* __restrict__ m2,
    const float* __restrict__ m3, const float* __restrict__ m4,
    float* __restrict__ P, float* __restrict__ chunkSum)
{
    const int mIdx  = blockIdx.y;
    const int chunk = blockIdx.x;
    const int j = threadIdx.x;
    const int c = j >> 5;
    const int d = j & 31;

    const float* mods[4] = { m1, m2, m3, m4 };
    const float* Km = K + (size_t)mIdx * T * D;
    const float* Vm = mods[mIdx];

    float acc = 0.0f;
    const int s0 = chunk * CS;
    for (int s = 0; s < CS; ++s) {
        const int si = s0 + s;
        acc += Km[(size_t)si * D + d] * Vm[(size_t)si * D + c];
        P[((size_t)mIdx * T + si) * 64 + j] = acc;
    }
    chunkSum[((size_t)mIdx * NCHUNK + chunk) * 64 + j] = acc;
}

// ---------------------------------------------------------------------------
// Kernel 3: exclusive scan of chunk totals (64 chunks, per modality).
// ---------------------------------------------------------------------------
__global__ __launch_bounds__(64) void scan_offsets_kernel(
    const float* __restrict__ chunkSum, float* __restrict__ chunkOff)
{
    const int mIdx = blockIdx.x;
    const int j = threadIdx.x;
    float run = 0.0f;
    for (int ch = 0; ch < NCHUNK; ++ch) {
        const size_t idx = ((size_t)mIdx * NCHUNK + ch) * 64 + j;
        const float v = chunkSum[idx];
        chunkOff[idx] = run;
        run += v;
    }
}

// ---------------------------------------------------------------------------
// Kernel 4: per query t: binary-search prefix length in each modality's
// sorted time channel, reconstruct A(n) = chunkOff + localPrefix, dot with Q.
// ---------------------------------------------------------------------------
__global__ __launch_bounds__(256) void finalize_kernel(
    const float* __restrict__ Q, const float* __restrict__ P,
    const float* __restrict__ chunkOff,
    const float* __restrict__ m1, const float* __restrict__ m2,
    const float* __restrict__ m3, const float* __restrict__ m4,
    float* __restrict__ out)
{
    const int t = blockIdx.x * blockDim.x + threadIdx.x;
    if (t >= T) return;

    const float t1v = m1[(size_t)t * D + (D - 1)];

    float q[D];
    const float4* q4 = (const float4*)(Q + (size_t)t * D);
    #pragma unroll
    for (int i = 0; i < D / 4; ++i) {
        const float4 v = q4[i];
        q[4*i+0] = v.x; q[4*i+1] = v.y; q[4*i+2] = v.z; q[4*i+3] = v.w;
    }

    const float* mods[4] = { m1, m2, m3, m4 };
    float z0 = 0.0f, z1 = 0.0f;

    for (int mIdx = 0; mIdx < 4; ++mIdx) {
        const float* Vm = mods[mIdx];
        // n = #{ s : t2[s] <= t1[t] } via upper_bound on sorted time channel
        int lo = 0, hi = T;
        while (lo < hi) {
            const int mid = (lo + hi) >> 1;
            if (Vm[(size_t)mid * D + (D - 1)] <= t1v) lo = mid + 1;
            else hi = mid;
        }
        const int n = lo;
        if (n > 0) {
            const int sIdx = n - 1;
            const int ch = sIdx >> 7;  // CS = 128
            const float* Pm = P + ((size_t)mIdx * T + sIdx) * 64;
            const float* Om = chunkOff + ((size_t)mIdx * NCHUNK + ch) * 64;
            #pragma unroll
            for (int d = 0; d < D; ++d) {
                const float A0 = Om[d]      + Pm[d];
                const float A1 = Om[32 + d] + Pm[32 + d];
                z0 += q[d] * A0;
                z1 += q[d] * A1;
            }
        }
    }
    out[(size_t)t * 2 + 0] = z0;
    out[(size_t)t * 2 + 1] = z1;
}

// ---------------------------------------------------------------------------
extern "C" void kernel_launch(void* const* d_in, const int* in_sizes, int n_in,
                              void* d_out, int out_size, void* d_ws, size_t ws_size,
                              hipStream_t stream) {
    const float* m1 = (const float*)d_in[0];
    const float* m2 = (const float*)d_in[1];
    const float* m3 = (const float*)d_in[2];
    const float* m4 = (const float*)d_in[3];
    const float* Wq = (const float*)d_in[4];
    const float* bq = (const float*)d_in[5];
    const float* Wk = (const float*)d_in[6];
    const float* bk = (const float*)d_in[7];
    float* out = (float*)d_out;

    // Workspace layout (floats): Q | K(4) | P(4*T*64) | chunkOff | chunkSum
    float* ws   = (float*)d_ws;
    float* Q    = ws;                              // T*D
    float* K    = Q + (size_t)T * D;               // 4*T*D
    float* P    = K + (size_t)4 * T * D;           // 4*T*64
    float* cOff = P + (size_t)4 * T * 64;          // 4*NCHUNK*64
    float* cSum = cOff + (size_t)4 * NCHUNK * 64;  // 4*NCHUNK*64

    // 1) Q and K_m via WMMA MLPs: 512 tiles x 5 jobs, one wave each.
    mlp_wmma_kernel<<<dim3(T / 16, 5), 32, 0, stream>>>(
        m1, m2, m3, m4, Wq, bq, Wk, bk, Q, K);

    // 2) Chunked prefix scan of K[s] (x) V[s,0:2].
    scan_chunk_kernel<<<dim3(NCHUNK, 4), 64, 0, stream>>>(
        K, m1, m2, m3, m4, P, cSum);

    // 3) Exclusive scan over chunk totals.
    scan_offsets_kernel<<<4, 64, 0, stream>>>(cSum, cOff);

    // 4) Binary search + Q-dot, write yhat[T,2].
    finalize_kernel<<<(T + 255) / 256, 256, 0, stream>>>(
        Q, P, cOff, m1, m2, m3, m4, out);
}